// Attention_7954279432380
// MI455X (gfx1250) — compile-verified
//
#include <hip/hip_runtime.h>
#include <hip/hip_bf16.h>

typedef __attribute__((ext_vector_type(2))) float v2f;
typedef __attribute__((ext_vector_type(8))) float v8f;
typedef __attribute__((ext_vector_type(4))) float f4;

#define B_    8
#define LTOT  4096
#define D_    512
#define H_    16
#define N_    256                 // LTOT / H_
#define ROWSTRIDE   (N_ * D_)     // floats between consecutive l-rows = 131072
#define BATCHSTRIDE (LTOT * D_)   // floats per batch = 2097152
#define INV_N 0.00390625f         // 1/256 exactly

// ---------------------------------------------------------------------------
// A-output is exactly 1/256 everywhere (softmax of a constant vector).
// 8*256*16*256 = 8388608 floats -> 2097152 float4 stores.
// ---------------------------------------------------------------------------
__global__ __launch_bounds__(256) void fill_A_kernel(float* __restrict__ a) {
    int i = blockIdx.x * blockDim.x + threadIdx.x;
    f4 val = {INV_N, INV_N, INV_N, INV_N};
    ((f4*)a)[i] = val;
}

// ---------------------------------------------------------------------------
// V[b, n*16+h, d] = (1/256) * sum_{l=0..15} v[b, l*256+n, d]   (same for all h)
//
// One wave handles one (b, n, d-half). The 16-way l-reduction is done with
// V_WMMA_F32_16X16X4_F32: A = ones(16x4), B = 4(l) x 16(d) tile of v,
// accumulated over 4 k-steps. All-ones A makes the result invariant to the
// exact K<->register mapping of the B layout; only lane&15 -> column matters.
// Every row of D equals the sum, so c[0] in every lane is the answer for
// column d_base + (lane&15).
// ---------------------------------------------------------------------------
__global__ __launch_bounds__(256) void v_reduce_wmma_kernel(
        const float* __restrict__ v, float* __restrict__ out) {
    const int lane  = threadIdx.x & 31;
    const int wib   = threadIdx.x >> 5;
    const int w     = blockIdx.x * 8 + wib;   // 4096 waves total
    const int dhalf = w & 1;
    const int bn    = w >> 1;
    const int b     = bn >> 8;                // / 256
    const int n     = bn & 255;
    const int col   = lane & 15;
    const int khalf = lane >> 4;              // 0: lanes 0-15, 1: lanes 16-31

    const float* vb = v   + (size_t)b * BATCHSTRIDE + (size_t)n * D_;
    float*       ob = out + (size_t)b * BATCHSTRIDE + (size_t)(n * H_) * D_;

    const v2f ones = {1.0f, 1.0f};

    for (int chunk = 0; chunk < 16; ++chunk) {
        const int dbase = dhalf * 256 + chunk * 16;
        const float* p = vb + dbase + col;

        v8f acc = {};
#pragma unroll
        for (int step = 0; step < 4; ++step) {
            // This wave-half supplies l-rows {l0, l0+1}; bijection onto l=0..15
            // across (step, khalf, reg) — order irrelevant under all-ones A.
            const int l0 = step * 4 + khalf * 2;
            v2f bm;
            bm.x = p[(size_t)l0 * ROWSTRIDE];
            bm.y = p[(size_t)(l0 + 1) * ROWSTRIDE];
            // 8 args: (neg_a, A, neg_b, B, c_mod, C, reuse_a, reuse_b)
            acc = __builtin_amdgcn_wmma_f32_16x16x4_f32(
                      false, ones, false, bm, (short)0, acc, false, false);
        }

        const float s = acc[0] * INV_N;   // identical across both lane halves

        // Broadcast to all 16 h rows: each store instr covers 2 rows
        // (lanes 0-15 -> row 2*it, lanes 16-31 -> row 2*it+1), 64B each.
#pragma unroll
        for (int it = 0; it < 8; ++it) {
            const int h = it * 2 + khalf;
            ob[(size_t)h * D_ + dbase + col] = s;
        }
    }
}

extern "C" void kernel_launch(void* const* d_in, const int* in_sizes, int n_in,
                              void* d_out, int out_size, void* d_ws, size_t ws_size,
                              hipStream_t stream) {
    // d_in[0] = q (mathematically unused: A is constant regardless of q)
    const float* v = (const float*)d_in[1];
    float* out = (float*)d_out;                       // V: 8*4096*512 floats
    float* Aout = out + (size_t)B_ * LTOT * D_;       // A: 8*256*16*256 floats

    const int a_count4 = (B_ * N_ * H_ * N_) / 4;     // 2097152 float4s
    fill_A_kernel<<<a_count4 / 256, 256, 0, stream>>>(Aout);

    // 4096 waves @ 8 waves (256 threads) per block -> 512 blocks
    v_reduce_wmma_kernel<<<512, 256, 0, stream>>>(v, out);
}